// UnaryDiv_55963423867501
// MI455X (gfx1250) — compile-verified
//
#include <hip/hip_runtime.h>
#include <cstdint>

// v4f as a true vector type so __builtin_nontemporal_store works on 16B.
typedef float v4f __attribute__((ext_vector_type(4)));

#define TPB   256          // 8 waves (wave32)
#define TILES 4            // tiles per block (double-buffered pipeline)
// elements per tile = TPB * 4 floats = 1024

__device__ __forceinline__ void async_ld16(unsigned lds_off, unsigned goff, uint64_t base) {
    // CDNA5 async data mover: memory -> LDS, 16B per lane, tracked by ASYNCcnt.
    // GVS addressing: mem = SGPR64 base + VGPR32 offset.
    asm volatile("global_load_async_to_lds_b128 %0, %1, %2"
                 :: "v"(lds_off), "v"(goff), "s"(base)
                 : "memory");
}

__global__ __launch_bounds__(TPB) void unary_div_bulk(
    const float* __restrict__ dividend,
    const float* __restrict__ divisor,
    const float* __restrict__ cnt,
    const float* __restrict__ sr,
    const int*   __restrict__ idxp,
    float*       __restrict__ out,
    long long srStride)
{
    __shared__ v4f buf[2][4][TPB];   // [phase][stream][thread] = 32 KB

    const int tid = threadIdx.x;
    const unsigned blk = blockIdx.x;

    // RNG_SEQ = {0,2,3,1}; tap = RNG_SEQ[idx % 4] packed as nibbles of 0x1320.
    const int idx = idxp[0];
    const int tap = (0x1320 >> ((idx & 3) * 4)) & 0xF;

    const uint64_t bA = (uint64_t)dividend;
    const uint64_t bB = (uint64_t)divisor;
    const uint64_t bC = (uint64_t)cnt;
    const uint64_t bH = (uint64_t)(sr + (long long)tap * srStride);

    // Per-lane LDS byte addresses (low 32 bits of generic pointer = LDS offset).
    unsigned ldsA[2], ldsB[2], ldsC[2], ldsH[2];
    #pragma unroll
    for (int p = 0; p < 2; ++p) {
        ldsA[p] = (unsigned)(uintptr_t)&buf[p][0][tid];
        ldsB[p] = (unsigned)(uintptr_t)&buf[p][1][tid];
        ldsC[p] = (unsigned)(uintptr_t)&buf[p][2][tid];
        ldsH[p] = (unsigned)(uintptr_t)&buf[p][3][tid];
    }

    auto stage = [&](unsigned t, int phase) {
        unsigned goff = ((blk * TILES + t) * TPB + (unsigned)tid) * 16u; // bytes
        async_ld16(ldsA[phase], goff, bA);
        async_ld16(ldsB[phase], goff, bB);
        async_ld16(ldsC[phase], goff, bC);
        async_ld16(ldsH[phase], goff, bH);
    };

    stage(0u, 0);
    #pragma unroll
    for (unsigned t = 0; t < TILES; ++t) {
        const int phase = (int)(t & 1u);
        if (t + 1 < TILES) {
            stage(t + 1, phase ^ 1);
            // 4 ops of tile t+1 outstanding; <=4 means tile t fully landed in LDS.
            asm volatile("s_wait_asynccnt 4" ::: "memory");
        } else {
            asm volatile("s_wait_asynccnt 0" ::: "memory");
        }

        const v4f a = buf[phase][0][tid];
        const v4f b = buf[phase][1][tid];
        const v4f c = buf[phase][2][tid];
        const v4f h = buf[phase][3][tid];

        v4f q;
        #pragma unroll
        for (int i = 0; i < 4; ++i) {
            const float ai = a[i], bi = b[i], ci = c[i], hi = h[i];
            // SkewedSync
            const float only_one = (ai + bi == 1.0f) ? 1.0f : 0.0f;
            const float cnt_not_min = (ci != 0.0f) ? 1.0f : 0.0f;
            const float cnt_not_max = (ci != 3.0f) ? 1.0f : 0.0f;
            const float sel = bi * cnt_not_min + (1.0f - bi) * cnt_not_max;
            const float dsync = ai + only_one * (2.0f * bi - 1.0f) * sel;
            // CORDIV kernel
            q[i] = (bi == 1.0f) ? dsync : hi;
        }

        const size_t gi = (size_t)(blk * TILES + t) * TPB + (size_t)tid;
        __builtin_nontemporal_store(q, (v4f*)out + gi);   // streaming store (th:NT)
    }
}

// Scalar tail for any N not divisible by TILES*TPB*4 (empty for the given shape).
__global__ void unary_div_tail(
    const float* __restrict__ dividend,
    const float* __restrict__ divisor,
    const float* __restrict__ cnt,
    const float* __restrict__ sr,
    const int*   __restrict__ idxp,
    float*       __restrict__ out,
    long long srStride, int start, int n)
{
    const int i = start + blockIdx.x * blockDim.x + threadIdx.x;
    if (i >= n) return;
    const int idx = idxp[0];
    const int tap = (0x1320 >> ((idx & 3) * 4)) & 0xF;
    const float ai = dividend[i], bi = divisor[i], ci = cnt[i];
    const float hi = sr[(long long)tap * srStride + i];
    const float only_one = (ai + bi == 1.0f) ? 1.0f : 0.0f;
    const float sel = bi * ((ci != 0.0f) ? 1.0f : 0.0f)
                    + (1.0f - bi) * ((ci != 3.0f) ? 1.0f : 0.0f);
    const float dsync = ai + only_one * (2.0f * bi - 1.0f) * sel;
    out[i] = (bi == 1.0f) ? dsync : hi;
}

extern "C" void kernel_launch(void* const* d_in, const int* in_sizes, int n_in,
                              void* d_out, int out_size, void* d_ws, size_t ws_size,
                              hipStream_t stream) {
    const float* dividend = (const float*)d_in[0];
    const float* divisor  = (const float*)d_in[1];
    const float* cnt      = (const float*)d_in[2];
    const float* sr       = (const float*)d_in[3];
    const int*   idxp     = (const int*)d_in[4];
    float*       out      = (float*)d_out;

    const int N = in_sizes[0];               // 16*1024*1024
    const long long srStride = N;            // sr is [4, N]

    const int elemsPerTile  = TPB * 4;       // 1024 floats
    const int elemsPerBlock = elemsPerTile * TILES;
    const int blocks = N / elemsPerBlock;    // 4096 for the given shape
    const int bulk   = blocks * elemsPerBlock;

    if (blocks > 0) {
        unary_div_bulk<<<blocks, TPB, 0, stream>>>(
            dividend, divisor, cnt, sr, idxp, out, srStride);
    }
    const int rem = N - bulk;
    if (rem > 0) {
        unary_div_tail<<<(rem + 255) / 256, 256, 0, stream>>>(
            dividend, divisor, cnt, sr, idxp, out, srStride, bulk, N);
    }
}